// GraphFeatureTokenizer_34926674051529
// MI455X (gfx1250) — compile-verified
//
#include <hip/hip_runtime.h>

// Problem constants from the reference
#define BTOT   16
#define NPER   2048
#define EPER   8192
#define DDIM   256
#define TWOD   512
#define TTOK   (NPER + EPER + 1)   // 10241
#define NNODES 50000

typedef __attribute__((ext_vector_type(2))) float v2f;
typedef __attribute__((ext_vector_type(4))) float v4f;
typedef __attribute__((ext_vector_type(8))) float v8f;

// ---------------------------------------------------------------------------
// K0: repack w_orf (D x 2D, row-major) into a WMMA-friendly B operand.
// Logical B: Bmat[c][j], c in [0,256) (K), j in [0,512) (N):
//   j <  256 : Bmat[c][j] = W[j][c]            (P half)
//   j >= 256 : Bmat[c][j] = W[j-256][256 + c]  (Q half)
// Stored K-pair interleaved: bprep[((c>>1)*512 + j)*2 + (c&1)]
// so a B-fragment (Bmat[kb][j], Bmat[kb+1][j]) is one aligned b64 load.
// ---------------------------------------------------------------------------
__global__ __launch_bounds__(256) void build_bprep(const float* __restrict__ w,
                                                   float* __restrict__ bprep) {
  int idx = blockIdx.x * 256 + threadIdx.x;   // 0 .. 256*512-1
  int c = idx >> 9;
  int j = idx & 511;
  float v = (j < DDIM) ? w[(size_t)j * TWOD + c]
                       : w[(size_t)(j - DDIM) * TWOD + DDIM + c];
  bprep[(((size_t)(c >> 1) * TWOD + j) << 1) + (c & 1)] = v;
}

// ---------------------------------------------------------------------------
// K1: PQ[m][0:512] = orf0[indices[m]] @ Bmat   (fp32 WMMA 16x16x4, K=256)
// Workgroup: 256 threads (8 waves), one 32-row M-block, full N=512.
// Wave w -> N columns [w*64, w*64+64), two 16-row M-tiles.
// ---------------------------------------------------------------------------
#define LDSA_STRIDE 260   // 256 + 4 pad: A-frag reads hit banks 4r+{0..3} => conflict-free

__global__ __launch_bounds__(256) void pq_gemm(const float* __restrict__ orf0,
                                               const int*   __restrict__ indices,
                                               const float* __restrict__ bprep,
                                               float*       __restrict__ pq) {
  __shared__ float ldsA[32 * LDSA_STRIDE];   // 33,280 bytes

  const int tid   = threadIdx.x;
  const int lane  = tid & 31;
  const int wv    = tid >> 5;          // 0..7
  const int mblk  = blockIdx.x;        // 32 rows per block
  const int r     = lane & 15;
  const int hi    = lane >> 4;         // 0 / 1
  const int koff  = hi << 1;           // 0 or 2 (K sub-offset within WMMA step)

  // ---- Stage gathered A tile (32 rows x 256) into LDS, coalesced b128 ----
#pragma unroll
  for (int i = 0; i < 8; ++i) {
    int p   = tid + i * 256;           // float4 index, 0..2047
    int row = p >> 6;                  // 64 float4 per row
    int k4  = p & 63;
    int g   = indices[mblk * 32 + row];
    v4f val = *reinterpret_cast<const v4f*>(orf0 + (size_t)g * DDIM + (k4 << 2));
    *reinterpret_cast<v4f*>(&ldsA[row * LDSA_STRIDE + (k4 << 2)]) = val;
  }
  __syncthreads();

  v8f acc[2][4];
#pragma unroll
  for (int mt = 0; mt < 2; ++mt)
#pragma unroll
    for (int t = 0; t < 4; ++t) {
      v8f z = {0.f, 0.f, 0.f, 0.f, 0.f, 0.f, 0.f, 0.f};
      acc[mt][t] = z;
    }

  // ---- K loop: 64 steps of K=4, 8 WMMA per step per wave ----
#pragma unroll 4
  for (int ks = 0; ks < 64; ++ks) {
    const int kabs = (ks << 2) + koff;                    // even
    v2f a0 = *reinterpret_cast<const v2f*>(&ldsA[r * LDSA_STRIDE + kabs]);
    v2f a1 = *reinterpret_cast<const v2f*>(&ldsA[(16 + r) * LDSA_STRIDE + kabs]);
#pragma unroll
    for (int t = 0; t < 4; ++t) {
      const int j = wv * 64 + t * 16 + r;
      v2f bf = *reinterpret_cast<const v2f*>(
          bprep + ((((size_t)kabs >> 1) * TWOD + j) << 1));
      acc[0][t] = __builtin_amdgcn_wmma_f32_16x16x4_f32(
          false, a0, false, bf, (short)0, acc[0][t], false, false);
      acc[1][t] = __builtin_amdgcn_wmma_f32_16x16x4_f32(
          false, a1, false, bf, (short)0, acc[1][t], false, false);
    }
  }

  // ---- Epilogue: C/D layout -> lane L holds (M = k + 8*hi, N = r) ----
#pragma unroll
  for (int mt = 0; mt < 2; ++mt)
#pragma unroll
    for (int t = 0; t < 4; ++t) {
      const int col = wv * 64 + t * 16 + r;
#pragma unroll
      for (int k = 0; k < 8; ++k) {
        const int m = mblk * 32 + mt * 16 + k + 8 * hi;
        pq[(size_t)m * TWOD + col] = acc[mt][t][k];
      }
    }
}

// ---------------------------------------------------------------------------
// K2: per-token assembly. One wave per token; lane handles 8 columns (b128).
// out tuple layout (all as float, concatenated in return order):
//   [B][T][D] out, [B][T] padding_mask, [B][T][2] padded_index,
//   [B][T] node_mask, [B][T] edge_mask
// ---------------------------------------------------------------------------
__global__ __launch_bounds__(256) void token_kernel(
    const float* __restrict__ node_data, const float* __restrict__ edge_data,
    const int*   __restrict__ edge_index, const float* __restrict__ order_w,
    const float* __restrict__ pq, float* __restrict__ out, int b0, int nb) {
  const long long totTok = (long long)nb * TTOK;
  const long long wid = (long long)blockIdx.x * 8 + (threadIdx.x >> 5);
  if (wid >= totTok) return;
  const int lane = threadIdx.x & 31;
  const int brel = (int)(wid / TTOK);
  const int t    = (int)(wid % TTOK);
  const int b    = b0 + brel;

  float* mpad  = out  + (size_t)BTOT * TTOK * DDIM;
  float* pidx  = mpad + (size_t)BTOT * TTOK;
  float* mnode = pidx + (size_t)BTOT * TTOK * 2;
  float* medge = mnode + (size_t)BTOT * TTOK;

  const size_t tokBase = (size_t)b * TTOK + t;
  float* orow = out + tokBase * DDIM;
  const int d0 = lane * 8;

  if (t == NPER + EPER) {                       // pad token
    v4f z = {0.f, 0.f, 0.f, 0.f};
    *reinterpret_cast<v4f*>(orow + d0)     = z;
    *reinterpret_cast<v4f*>(orow + d0 + 4) = z;
    if (lane == 0) {
      mpad[tokBase]      = 1.f;
      pidx[2 * tokBase]     = 0.f;
      pidx[2 * tokBase + 1] = 0.f;
      mnode[tokBase]     = 0.f;
      medge[tokBase]     = 0.f;
    }
    return;
  }

  int i0, i1, order;
  const float* feat;
  const int isNode = (t < NPER);
  if (isNode) {
    i0 = i1 = t;
    feat = node_data + ((size_t)b * NPER + t) * DDIM;
    order = 1;
  } else {
    const int j = t - NPER;
    const size_t eoff = (size_t)b * EPER + j;
    i0 = edge_index[eoff];
    i1 = edge_index[(size_t)BTOT * EPER + eoff];
    feat = edge_data + eoff * DDIM;
    order = (i0 == i1) ? 1 : 0;
  }

  const float* prow = pq + ((size_t)brel * NPER + i0) * TWOD + d0;          // P half
  const float* qrow = pq + ((size_t)brel * NPER + i1) * TWOD + DDIM + d0;   // Q half
  const float* ow   = order_w + (size_t)order * DDIM + d0;

#pragma unroll
  for (int h = 0; h < 2; ++h) {
    v4f f = *reinterpret_cast<const v4f*>(feat + d0 + h * 4);
    v4f p = *reinterpret_cast<const v4f*>(prow + h * 4);
    v4f q = *reinterpret_cast<const v4f*>(qrow + h * 4);
    v4f w = *reinterpret_cast<const v4f*>(ow + h * 4);
    v4f o = f + p + q + w;
    *reinterpret_cast<v4f*>(orow + d0 + h * 4) = o;
  }

  if (lane == 0) {
    mpad[tokBase]         = 0.f;
    pidx[2 * tokBase]     = (float)i0;
    pidx[2 * tokBase + 1] = (float)i1;
    mnode[tokBase]        = isNode ? 1.f : 0.f;
    medge[tokBase]        = isNode ? 0.f : 1.f;
  }
}

// ---------------------------------------------------------------------------
extern "C" void kernel_launch(void* const* d_in, const int* in_sizes, int n_in,
                              void* d_out, int out_size, void* d_ws, size_t ws_size,
                              hipStream_t stream) {
  (void)in_sizes; (void)n_in; (void)out_size;
  const float* node_data  = (const float*)d_in[0];
  const float* edge_data  = (const float*)d_in[1];
  const float* orf0       = (const float*)d_in[2];
  const float* w_orf      = (const float*)d_in[3];
  const float* order_w    = (const float*)d_in[4];
  const int*   edge_index = (const int*)d_in[5];
  const int*   indices    = (const int*)d_in[6];
  // d_in[7]=node_num, d_in[8]=edge_num: compile-time constants (2048 / 8192)

  float* out   = (float*)d_out;
  float* bprep = (float*)d_ws;
  const size_t bprepBytes = (size_t)DDIM * TWOD * sizeof(float);  // 512 KB
  float* pqbuf = (float*)((char*)d_ws + bprepBytes);

  // K0: repack weights (512 blocks x 256 threads covers 256*512 elements)
  build_bprep<<<512, 256, 0, stream>>>(w_orf, bprep);

  // Batch-group size from available scratch (4 MB of PQ per batch).
  const size_t perBatch = (size_t)NPER * TWOD * sizeof(float);
  size_t avail = (ws_size > bprepBytes) ? (ws_size - bprepBytes) : 0;
  int G = (int)(avail / perBatch);
  if (G < 1) G = 1;
  if (G > BTOT) G = BTOT;

  for (int bstart = 0; bstart < BTOT; bstart += G) {
    const int nb = (BTOT - bstart < G) ? (BTOT - bstart) : G;
    // K1: gathered node transform -> PQ rows for this batch group
    pq_gemm<<<nb * (NPER / 32), 256, 0, stream>>>(
        orf0, indices + (size_t)bstart * NPER, bprep, pqbuf);
    // K2: token assembly for this batch group
    const long long tok = (long long)nb * TTOK;
    const int blocks = (int)((tok + 7) / 8);
    token_kernel<<<blocks, 256, 0, stream>>>(
        node_data, edge_data, edge_index, order_w, pqbuf, out, bstart, nb);
  }
}